// AdapterMACE_61074434949134
// MI455X (gfx1250) — compile-verified
//
#include <hip/hip_runtime.h>
#include <math.h>

// ============================================================================
// MACE forward for MI455X (gfx1250), wave32.
//  - All dense contractions: V_WMMA_F32_16X16X4_F32 (full f32, matches ref).
//  - Weight panels staged into LDS once per 8-wave workgroup via the Tensor
//    Data Mover (tensor_load_to_lds + s_wait_tensorcnt), then read as ds_load.
//  - Edge scatter-sum via f32 global atomics, radial MLP chunked over edges.
//
// Input flattening assumed: recursive dict-insertion order of setup_inputs():
//  0 coordinates[N,3] 1 node_attrs[N,10] 2 edge_index[2,E](int)
//  3 node_embedding[10,64]
//  layer0: 4 up0  5 rad.w1[8,64] 6 rad.b1 7 rad.w2 8 rad.b2 9 rad.w3[64,128]
//          10 lin0 11 lin1 12 skip0[10,64,64] 13 skip1
//          14 msg.w1 15 msg.w2 16 msg.w3 17 msg.v1 18 msg.v2 19 msg.lin0 20 msg.lin1
//          21 nu.lin0 22 nu.lin1
//          23 bn.w0 24 bn.b0 25 bn.w1
//          26 ab.d0[64,16] 27 ab.u0[16,64] 28 ab.bn.w0 29 ab.bn.b0 30 ab.bn.w1
//          31 aa.d0 32 aa.d1 33 aa.u0 34 aa.u1 35 aa.bn.w0 36 aa.bn.b0 37 aa.bn.w1
//          38 s_before 39 s_after
//  layer1: 40 up0 41 up1 42 rad.w1 43 rad.b1 44 rad.w2 45 rad.b2 46 rad.w3[64,256]
//          47 lin0 48 lin1
//          49 msg.w1 50 msg.w2 51 msg.w3 52 msg.lin0
//          53 nu.lin0 54 nu.skip0[10,64,64]
//          55 bn.w0 56 bn.b0
//          57 ab.d0 58 ab.u0 59 ab.bn.w0 60 ab.bn.b0
//          61 aa.d0 62 aa.u0 63 aa.bn.w0 64 aa.bn.b0
//          65 s_before 66 s_after
// ============================================================================

typedef float v2f __attribute__((ext_vector_type(2)));
typedef float v8f __attribute__((ext_vector_type(8)));
typedef unsigned int u32x4 __attribute__((ext_vector_type(4)));
typedef int i32x4 __attribute__((ext_vector_type(4)));
typedef int i32x8 __attribute__((ext_vector_type(8)));

#define MUL 64
#define NZ 10
#define EPSBN 1e-5f
#define INV_SQRT3 0.57735026918962576f
#define INV_SQRT2 0.70710678118654752f
#define SQRT3F 1.7320508075688772f

__device__ __forceinline__ float silu_f(float v) {
    return v / (1.0f + __expf(-v));
}

// ---------------------------------------------------------------------------
// TDM stage: async-copy the B weight panel [K,N] (row-major f32, contiguous)
// into LDS offset 0 (base of dynamic LDS). Issued by one wave; caller must
// __syncthreads() afterwards. Descriptor per CDNA5 ISA 8.3-8.5:
//   group0: count=1 | lds_addr | global_addr[56:0] | type=2
//   group1: data_size=4B, tensor_dim0=N, tensor_dim1=K, tile_dim0=N,
//           tile_dim1=K, tensor_dim0_stride=N
// This toolchain (clang-23 / therock-10.0) exposes the 6-arg builtin:
//   (u32x4 g0, i32x8 g1, i32x4 g2, i32x4 g3, i32x8 g4, i32 cpol)
// ---------------------------------------------------------------------------
__device__ __forceinline__ void tdm_stage_B(const float* B, int K, int N)
{
    unsigned long long ga = (unsigned long long)(size_t)B;
    u32x4 g0;
    g0[0] = 1u;                                    // count=1 (valid), user mode
    g0[1] = 0u;                                    // lds_addr = 0 (dyn-LDS base)
    g0[2] = (unsigned)(ga & 0xFFFFFFFFu);          // global_addr[31:0]
    g0[3] = (unsigned)((ga >> 32) & 0x01FFFFFFu)   // global_addr[56:32]
          | (2u << 30);                            // type=2 ("image")
    i32x8 g1;
    g1[0] = (int)(2u << 16);                       // data_size=2 -> 4 bytes
    g1[1] = (int)(((unsigned)N & 0xFFFFu) << 16);  // tensor_dim0[15:0]
    g1[2] = (int)((((unsigned)K & 0xFFFFu) << 16)  // tensor_dim1[15:0]
          | (((unsigned)N >> 16) & 0xFFFFu));      // tensor_dim0[31:16]
    g1[3] = (int)(((unsigned)N & 0xFFFFu) << 16);  // tile_dim0 = N
    g1[4] = (int)((unsigned)K & 0xFFFFu);          // tile_dim1 = K, tile_dim2=0
    g1[5] = (int)(unsigned)N;                      // tensor_dim0_stride[31:0]
    g1[6] = 0;                                     // stride hi / dim1_stride lo
    g1[7] = 0;
    i32x4 gz4 = {0, 0, 0, 0};
    i32x8 gz8 = {0, 0, 0, 0, 0, 0, 0, 0};
    __builtin_amdgcn_tensor_load_to_lds(g0, g1, gz4, gz4, gz8, 0);
    __builtin_amdgcn_s_wait_tensorcnt(0);
}

// ---------------------------------------------------------------------------
// WMMA GEMM: C[M,N] = act(alpha * A[M,K] @ B[K,N] + bias[N])
// 256-thread block = 8 waves; block covers 128 rows; wave w covers rows
// [tile*16, tile*16+16). B panel staged in LDS by the TDM, shared by all
// waves. M multiple of 16, K multiple of 4, N = NT*16.
// Fragment layouts per CDNA5 ISA 7.12.2:
//   A frag (16x4 f32): lane l<16 row=l holds K={0,1}; lanes>=16 hold K={2,3}
//   B frag (4x16 f32): lane l<16 col=l holds K={0,1}; lanes>=16 hold K={2,3}
//   C frag (16x16 f32): lane l<16 col=l rows 0..7 in v0..7; lanes>=16 rows 8..15
// ---------------------------------------------------------------------------
template <int NT>
__global__ __launch_bounds__(256) void k_gemm_wmma(
    const float* __restrict__ A, const float* __restrict__ B,
    float* __restrict__ C, const float* __restrict__ bias,
    int M, int K, float alpha, int act)
{
    extern __shared__ float Bs[];  // [K][N] at dynamic-LDS base (offset 0)
    const int N = NT * 16;

    if (threadIdx.x < 32) tdm_stage_B(B, K, N);  // one wave issues the DMA
    __syncthreads();

    const int tile = blockIdx.x * 8 + (threadIdx.x >> 5);
    if (tile * 16 >= M) return;
    const int m0 = tile * 16;
    const int lane = threadIdx.x & 31;
    const int half = lane >> 4;
    const int l = lane & 15;

    v8f acc[NT] = {};
    const float* arow = A + (size_t)(m0 + l) * K + 2 * half;
    for (int k = 0; k < K; k += 4) {
        v2f a;
        a.x = arow[k];
        a.y = arow[k + 1];
        const float* bp = Bs + (size_t)(k + 2 * half) * N + l;
#pragma unroll
        for (int t = 0; t < NT; ++t) {
            v2f b;
            b.x = bp[t * 16];
            b.y = bp[t * 16 + N];
            acc[t] = __builtin_amdgcn_wmma_f32_16x16x4_f32(
                false, a, false, b, (short)0, acc[t], false, false);
        }
    }
#pragma unroll
    for (int t = 0; t < NT; ++t) {
        const int n = t * 16 + l;
        const float bv = bias ? bias[n] : 0.0f;
#pragma unroll
        for (int i = 0; i < 8; ++i) {
            const int m = m0 + i + 8 * half;
            float v = alpha * acc[t][i] + bv;
            if (act) v = silu_f(v);
            C[(size_t)m * N + n] = v;
        }
    }
}

// ---------------------------------------------------------------------------
// Elementwise / gather / scatter kernels
// ---------------------------------------------------------------------------

__global__ void k_argmax_z(const float* __restrict__ attrs, int* __restrict__ z, int n)
{
    int i = blockIdx.x * blockDim.x + threadIdx.x;
    if (i >= n) return;
    const float* row = attrs + (size_t)i * NZ;
    int best = 0;
    float bv = row[0];
    for (int k = 1; k < NZ; ++k) {
        float v = row[k];
        if (v > bv) { bv = v; best = k; }
    }
    z[i] = best;
}

__global__ void k_embed(const float* __restrict__ emb, const int* __restrict__ z,
                        float* __restrict__ f0, int n)
{
    int idx = blockIdx.x * blockDim.x + threadIdx.x;
    if (idx >= n * MUL) return;
    int i = idx >> 6, j = idx & 63;
    f0[idx] = emb[z[i] * MUL + j];
}

__global__ void k_edge_geom(const float* __restrict__ coord, const int* __restrict__ ei,
                            float* __restrict__ Y1, float* __restrict__ ef, int E)
{
    int e = blockIdx.x * blockDim.x + threadIdx.x;
    if (e >= E) return;
    int s = ei[e];
    int r = ei[E + e];
    float dx = coord[r * 3 + 0] - coord[s * 3 + 0];
    float dy = coord[r * 3 + 1] - coord[s * 3 + 1];
    float dz = coord[r * 3 + 2] - coord[s * 3 + 2];
    float rr = sqrtf(dx * dx + dy * dy + dz * dz);
    float rs = fmaxf(rr, 1e-6f);
    float inv = 1.0f / rs;
    Y1[e * 3 + 0] = SQRT3F * dx * inv;
    Y1[e * 3 + 1] = SQRT3F * dy * inv;
    Y1[e * 3 + 2] = SQRT3F * dz * inv;
    // bessel * polynomial cutoff (R=5, p=5)
    float x = rs * 0.2f;
    float env = 0.0f;
    if (x < 1.0f) {
        float x2 = x * x;
        float x5 = x2 * x2 * x;
        env = 1.0f - 21.0f * x5 + 35.0f * x5 * x - 15.0f * x5 * x2;
    }
    float pref = sqrtf(0.4f) * inv * env;
    float base = (float)M_PI * 0.2f * rs;
#pragma unroll
    for (int nb = 1; nb <= 8; ++nb) {
        ef[(size_t)e * 8 + nb - 1] = pref * sinf(base * (float)nb);
    }
}

// layer0 message formation + scatter-sum (f1 absent)
__global__ void k_scatter_l0(const float* __restrict__ W /*[cnt,128]*/,
                             const float* __restrict__ x0 /*[N,64]*/,
                             const float* __restrict__ Y1 /*[E,3]*/,
                             const int* __restrict__ send, const int* __restrict__ recv,
                             float* __restrict__ seg0, float* __restrict__ seg1,
                             int eBase, int cnt)
{
    int idx = blockIdx.x * blockDim.x + threadIdx.x;
    if (idx >= cnt * MUL) return;
    int el = idx >> 6, k = idx & 63;
    int e = eBase + el;
    int s = send[e], r = recv[e];
    float xs = x0[(size_t)s * MUL + k];
    float w0 = W[(size_t)el * 128 + k];
    float w1 = W[(size_t)el * 128 + 64 + k];
    float y0 = Y1[e * 3 + 0], y1 = Y1[e * 3 + 1], y2 = Y1[e * 3 + 2];
    float c = w1 * xs;
    atomicAdd(&seg0[(size_t)r * 64 + k], w0 * xs);
    atomicAdd(&seg1[(size_t)r * 192 + 0 * 64 + k], c * y0);
    atomicAdd(&seg1[(size_t)r * 192 + 1 * 64 + k], c * y1);
    atomicAdd(&seg1[(size_t)r * 192 + 2 * 64 + k], c * y2);
}

// layer1 message formation + scatter-sum (with f1)
__global__ void k_scatter_l1(const float* __restrict__ W /*[cnt,256]*/,
                             const float* __restrict__ x0 /*[N,64]*/,
                             const float* __restrict__ x1 /*[N,3,64]*/,
                             const float* __restrict__ Y1,
                             const int* __restrict__ send, const int* __restrict__ recv,
                             float* __restrict__ seg0, float* __restrict__ seg1,
                             int eBase, int cnt)
{
    int idx = blockIdx.x * blockDim.x + threadIdx.x;
    if (idx >= cnt * MUL) return;
    int el = idx >> 6, k = idx & 63;
    int e = eBase + el;
    int s = send[e], r = recv[e];
    float xs0 = x0[(size_t)s * MUL + k];
    float xs1a = x1[(size_t)s * 192 + 0 * 64 + k];
    float xs1b = x1[(size_t)s * 192 + 1 * 64 + k];
    float xs1c = x1[(size_t)s * 192 + 2 * 64 + k];
    const float* wr = W + (size_t)el * 256;
    float w0 = wr[k], w1 = wr[64 + k], w2 = wr[128 + k], w3 = wr[192 + k];
    float y0 = Y1[e * 3 + 0], y1 = Y1[e * 3 + 1], y2 = Y1[e * 3 + 2];
    float dotY = xs1a * y0 + xs1b * y1 + xs1c * y2;
    float m0 = (w0 * xs0 + w1 * dotY * INV_SQRT3) * INV_SQRT2;
    float cw = w2 * xs0;
    atomicAdd(&seg0[(size_t)r * 64 + k], m0);
    atomicAdd(&seg1[(size_t)r * 192 + 0 * 64 + k], (cw * y0 + w3 * xs1a) * INV_SQRT2);
    atomicAdd(&seg1[(size_t)r * 192 + 1 * 64 + k], (cw * y1 + w3 * xs1b) * INV_SQRT2);
    atomicAdd(&seg1[(size_t)r * 192 + 2 * 64 + k], (cw * y2 + w3 * xs1c) * INV_SQRT2);
}

// out[row,:] (+)= X[row,:] @ Wz[z[row/rpn]]  where Wz is [NZ,64,64]
__global__ __launch_bounds__(64) void k_skip_gemv(
    const float* __restrict__ X, const float* __restrict__ Wz,
    const int* __restrict__ z, float* __restrict__ Out,
    int rowsPerNode, int accumulate)
{
    int row = blockIdx.x;
    int j = threadIdx.x;
    int node = row / rowsPerNode;
    const float* w = Wz + (size_t)z[node] * 4096 + j;
    const float* x = X + (size_t)row * 64;
    float acc = 0.0f;
#pragma unroll 8
    for (int k = 0; k < 64; ++k) acc += x[k] * w[(size_t)k * 64];
    if (accumulate) Out[(size_t)row * 64 + j] += acc;
    else Out[(size_t)row * 64 + j] = acc;
}

// symmetric contraction (correlation=2) elementwise part
__global__ void k_msg_elem(const float* __restrict__ a0, const float* __restrict__ a1,
                           const int* __restrict__ z,
                           const float* __restrict__ w1, const float* __restrict__ w2,
                           const float* __restrict__ w3,
                           const float* __restrict__ v1, const float* __restrict__ v2,
                           float* __restrict__ B0, float* __restrict__ B1, int n)
{
    int idx = blockIdx.x * blockDim.x + threadIdx.x;
    if (idx >= n * MUL) return;
    int i = idx >> 6, k = idx & 63;
    int zz = z[i];
    float a0v = a0[idx];
    float a = a1[(size_t)i * 192 + 0 * 64 + k];
    float b = a1[(size_t)i * 192 + 1 * 64 + k];
    float c = a1[(size_t)i * 192 + 2 * 64 + k];
    float nsq = a * a + b * b + c * c;
    B0[idx] = w1[zz * MUL + k] * a0v + w2[zz * MUL + k] * a0v * a0v
            + w3[zz * MUL + k] * nsq * INV_SQRT3;
    if (B1) {
        float s = v1[zz * MUL + k] + v2[zz * MUL + k] * a0v;
        B1[(size_t)i * 192 + 0 * 64 + k] = s * a;
        B1[(size_t)i * 192 + 1 * 64 + k] = s * b;
        B1[(size_t)i * 192 + 2 * 64 + k] = s * c;
    }
}

// BN stats over scalars: one block per channel j, stride-64 rows
__global__ __launch_bounds__(256) void k_bn_stats0(
    const float* __restrict__ x, int nrows, float* __restrict__ s1o, float* __restrict__ s2o)
{
    int j = blockIdx.x;
    float s1 = 0.0f, s2 = 0.0f;
    for (int r = threadIdx.x; r < nrows; r += 256) {
        float v = x[(size_t)r * 64 + j];
        s1 += v;
        s2 += v * v;
    }
    __shared__ float sh1[256], sh2[256];
    sh1[threadIdx.x] = s1;
    sh2[threadIdx.x] = s2;
    __syncthreads();
    for (int off = 128; off > 0; off >>= 1) {
        if (threadIdx.x < off) {
            sh1[threadIdx.x] += sh1[threadIdx.x + off];
            sh2[threadIdx.x] += sh2[threadIdx.x + off];
        }
        __syncthreads();
    }
    if (threadIdx.x == 0) { s1o[j] = sh1[0]; s2o[j] = sh2[0]; }
}

// BN stats for l=1 channels: sum over nodes of |x1[n,:,j]|^2
__global__ __launch_bounds__(256) void k_bn_stats1(
    const float* __restrict__ x1, int nrows, float* __restrict__ so)
{
    int j = blockIdx.x;
    float s = 0.0f;
    for (int r = threadIdx.x; r < nrows; r += 256) {
        float a = x1[(size_t)r * 192 + 0 * 64 + j];
        float b = x1[(size_t)r * 192 + 1 * 64 + j];
        float c = x1[(size_t)r * 192 + 2 * 64 + j];
        s += a * a + b * b + c * c;
    }
    __shared__ float sh[256];
    sh[threadIdx.x] = s;
    __syncthreads();
    for (int off = 128; off > 0; off >>= 1) {
        if (threadIdx.x < off) sh[threadIdx.x] += sh[threadIdx.x + off];
        __syncthreads();
    }
    if (threadIdx.x == 0) so[j] = sh[0];
}

__global__ void k_bn_apply0(const float* __restrict__ x, const float* __restrict__ w0,
                            const float* __restrict__ b0,
                            const float* __restrict__ s1, const float* __restrict__ s2,
                            float* __restrict__ y, int n)
{
    int idx = blockIdx.x * blockDim.x + threadIdx.x;
    if (idx >= n * MUL) return;
    int j = idx & 63;
    float invN = 1.0f / (float)n;
    float mu = s1[j] * invN;
    float var = s2[j] * invN - mu * mu;
    y[idx] = w0[j] * (x[idx] - mu) * rsqrtf(var + EPSBN) + b0[j];
}

__global__ void k_bn_apply1(const float* __restrict__ x1, const float* __restrict__ w1,
                            const float* __restrict__ sn2, float* __restrict__ y1, int n)
{
    int idx = blockIdx.x * blockDim.x + threadIdx.x;
    if (idx >= n * 192) return;
    int j = idx & 63;
    float n2m = sn2[j] / (float)n;
    y1[idx] = w1[j] * rsqrtf(n2m + EPSBN) * x1[idx];
}

// f0_new = b0 + sb*ab0 + sa*aa0 ; write to out[n, colOff + j] (ldout=320)
__global__ void k_combine0(const float* __restrict__ b0, const float* __restrict__ ab0,
                           const float* __restrict__ aa0,
                           const float* __restrict__ sb, const float* __restrict__ sa,
                           float* __restrict__ f0out, float* __restrict__ out,
                           int colOff, int n)
{
    int idx = blockIdx.x * blockDim.x + threadIdx.x;
    if (idx >= n * MUL) return;
    int i = idx >> 6, j = idx & 63;
    float v = b0[idx] + sb[0] * ab0[idx] + sa[0] * aa0[idx];
    if (f0out) f0out[idx] = v;
    out[(size_t)i * 320 + colOff + j] = v;
}

// f1_new[n,d,k] = b1 + sb*ab1 + sa*aa1 ; write out[n, 64 + k*3 + d]
__global__ void k_combine1(const float* __restrict__ b1, const float* __restrict__ ab1,
                           const float* __restrict__ aa1,
                           const float* __restrict__ sb, const float* __restrict__ sa,
                           float* __restrict__ f1out, float* __restrict__ out, int n)
{
    int idx = blockIdx.x * blockDim.x + threadIdx.x;
    if (idx >= n * 192) return;
    int i = idx / 192;
    int rem = idx - i * 192;
    int d = rem >> 6, k = rem & 63;
    float v = b1[idx] + sb[0] * ab1[idx] + sa[0] * aa1[idx];
    if (f1out) f1out[idx] = v;
    out[(size_t)i * 320 + 64 + k * 3 + d] = v;
}

// ---------------------------------------------------------------------------
// Host orchestration
// ---------------------------------------------------------------------------
static inline void gemm(const float* A, const float* B, float* C, const float* bias,
                        int M, int K, int N, float alpha, int act, hipStream_t s)
{
    int tiles = M / 16;
    dim3 g((tiles + 7) / 8), b(256);
    size_t shmem = (size_t)K * N * sizeof(float);
    switch (N) {
        case 16:  k_gemm_wmma<1><<<g, b, shmem, s>>>(A, B, C, bias, M, K, alpha, act); break;
        case 64:  k_gemm_wmma<4><<<g, b, shmem, s>>>(A, B, C, bias, M, K, alpha, act); break;
        case 128: k_gemm_wmma<8><<<g, b, shmem, s>>>(A, B, C, bias, M, K, alpha, act); break;
        case 256: k_gemm_wmma<16><<<g, b, shmem, s>>>(A, B, C, bias, M, K, alpha, act); break;
        default: break;
    }
}

extern "C" void kernel_launch(void* const* d_in, const int* in_sizes, int n_in,
                              void* d_out, int out_size, void* d_ws, size_t ws_size,
                              hipStream_t stream)
{
    if (n_in < 67) return;
    const int N = in_sizes[0] / 3;       // 30000
    const int E = in_sizes[2] / 2;       // 480000
    const int CH = E / 8;                // edge chunk

    const float* coord = (const float*)d_in[0];
    const float* attrs = (const float*)d_in[1];
    const int* eidx = (const int*)d_in[2];
    const int* send = eidx;
    const int* recv = eidx + E;
#define PIN(i) ((const float*)d_in[(i)])

    // workspace bump allocator (floats)
    size_t off = 0;
    auto alloc = [&](size_t cnt) {
        float* p = (float*)d_ws + off;
        off += cnt;
        return p;
    };
    int* z = (int*)alloc(N);
    float* Y1 = alloc((size_t)E * 3);
    float* ef = alloc((size_t)E * 8);
    float* f0a = alloc((size_t)N * 64);
    float* f1a = alloc((size_t)N * 192);  // placeholder (layer0 has no f1 input)
    float* f0b = alloc((size_t)N * 64);
    float* f1b = alloc((size_t)N * 192);
    float* x0up = alloc((size_t)N * 64);
    float* x1up = alloc((size_t)N * 192);
    float* seg0 = alloc((size_t)N * 64);   // also reused as u0
    float* seg1 = alloc((size_t)N * 192);  // also reused as u1
    float* a0 = alloc((size_t)N * 64);
    float* a1 = alloc((size_t)N * 192);
    float* t0 = alloc((size_t)N * 64);
    float* t1 = alloc((size_t)N * 192);
    float* ab0 = alloc((size_t)N * 64);
    float* ab1 = alloc((size_t)N * 192);
    float* aa0 = alloc((size_t)N * 64);
    float* aa1 = alloc((size_t)N * 192);
    float* m0b = alloc((size_t)N * 64);   // also reused as b0 (post-BN)
    float* m1b = alloc((size_t)N * 192);  // also reused as b1
    float* d16 = alloc((size_t)N * 16);
    float* d16v = alloc((size_t)N * 48);
    float* h1 = alloc((size_t)CH * 64);
    float* h2 = alloc((size_t)CH * 64);
    float* Wc = alloc((size_t)CH * 256);
    float* stats = alloc(192);  // s1[64], s2[64], n2[64]
    float* s1 = stats, * s2 = stats + 64, * sn2 = stats + 128;
    (void)f1a; (void)ws_size; (void)out_size;

    float* out = (float*)d_out;
    const int TB = 256;
    auto blocks = [&](size_t tot) { return dim3((unsigned)((tot + TB - 1) / TB)); };
    const float invAvg = 1.0f / 16.0f;

    // ---- setup ----
    k_argmax_z<<<blocks(N), TB, 0, stream>>>(attrs, z, N);
    k_edge_geom<<<blocks(E), TB, 0, stream>>>(coord, eidx, Y1, ef, E);
    k_embed<<<blocks((size_t)N * 64), TB, 0, stream>>>(PIN(3), z, f0a, N);
    (void)hipMemsetAsync(ab1, 0, (size_t)N * 192 * sizeof(float), stream);  // layer0 ab1 == 0

    // ======================= LAYER 0 =======================
    // adapter-before: f0a -> d16 -> ab0, BN in place; ab1 stays zero
    gemm(f0a, PIN(26), d16, nullptr, N, 64, 16, 1.0f, 1, stream);
    gemm(d16, PIN(27), ab0, nullptr, N, 16, 64, 1.0f, 0, stream);
    k_bn_stats0<<<64, 256, 0, stream>>>(ab0, N, s1, s2);
    k_bn_apply0<<<blocks((size_t)N * 64), TB, 0, stream>>>(ab0, PIN(28), PIN(29), s1, s2, ab0, N);

    // interaction
    gemm(f0a, PIN(4), x0up, nullptr, N, 64, 64, 1.0f, 0, stream);
    (void)hipMemsetAsync(seg0, 0, (size_t)N * 64 * sizeof(float), stream);
    (void)hipMemsetAsync(seg1, 0, (size_t)N * 192 * sizeof(float), stream);
    for (int c = 0; c < 8; ++c) {
        int base = c * CH;
        gemm(ef + (size_t)base * 8, PIN(5), h1, PIN(6), CH, 8, 64, 1.0f, 1, stream);
        gemm(h1, PIN(7), h2, PIN(8), CH, 64, 64, 1.0f, 1, stream);
        gemm(h2, PIN(9), Wc, nullptr, CH, 64, 128, 1.0f, 0, stream);
        k_scatter_l0<<<blocks((size_t)CH * 64), TB, 0, stream>>>(
            Wc, x0up, Y1, send, recv, seg0, seg1, base, CH);
    }
    gemm(seg0, PIN(10), t0, nullptr, N, 64, 64, invAvg, 0, stream);
    k_skip_gemv<<<N, 64, 0, stream>>>(t0, PIN(12), z, a0, 1, 0);
    gemm(seg1, PIN(11), t1, nullptr, 3 * N, 64, 64, invAvg, 0, stream);
    k_skip_gemv<<<3 * N, 64, 0, stream>>>(t1, PIN(13), z, a1, 3, 0);

    // message (symmetric contraction)
    k_msg_elem<<<blocks((size_t)N * 64), TB, 0, stream>>>(
        a0, a1, z, PIN(14), PIN(15), PIN(16), PIN(17), PIN(18), t0, t1, N);
    gemm(t0, PIN(19), m0b, nullptr, N, 64, 64, 1.0f, 0, stream);
    gemm(t1, PIN(20), m1b, nullptr, 3 * N, 64, 64, 1.0f, 0, stream);

    // node update (no residual in layer0): u0=seg0, u1=seg1
    gemm(m0b, PIN(21), seg0, nullptr, N, 64, 64, 1.0f, 0, stream);
    gemm(m1b, PIN(22), seg1, nullptr, 3 * N, 64, 64, 1.0f, 0, stream);

    // layer BN: b0 -> m0b, b1 -> m1b
    k_bn_stats0<<<64, 256, 0, stream>>>(seg0, N, s1, s2);
    k_bn_apply0<<<blocks((size_t)N * 64), TB, 0, stream>>>(seg0, PIN(23), PIN(24), s1, s2, m0b, N);
    k_bn_stats1<<<64, 256, 0, stream>>>(seg1, N, sn2);
    k_bn_apply1<<<blocks((size_t)N * 192), TB, 0, stream>>>(seg1, PIN(25), sn2, m1b, N);

    // adapter-after (with l=1): u0=seg0, u1=seg1
    gemm(seg0, PIN(31), d16, nullptr, N, 64, 16, 1.0f, 1, stream);
    gemm(d16, PIN(33), aa0, nullptr, N, 16, 64, 1.0f, 0, stream);
    gemm(seg1, PIN(32), d16v, nullptr, 3 * N, 64, 16, 1.0f, 0, stream);  // no SiLU on l=1
    gemm(d16v, PIN(34), aa1, nullptr, 3 * N, 16, 64, 1.0f, 0, stream);
    k_bn_stats0<<<64, 256, 0, stream>>>(aa0, N, s1, s2);
    k_bn_apply0<<<blocks((size_t)N * 64), TB, 0, stream>>>(aa0, PIN(35), PIN(36), s1, s2, aa0, N);
    k_bn_stats1<<<64, 256, 0, stream>>>(aa1, N, sn2);
    k_bn_apply1<<<blocks((size_t)N * 192), TB, 0, stream>>>(aa1, PIN(37), sn2, aa1, N);

    // combine -> f0b,f1b and output cols [0,256)
    k_combine0<<<blocks((size_t)N * 64), TB, 0, stream>>>(
        m0b, ab0, aa0, PIN(38), PIN(39), f0b, out, 0, N);
    k_combine1<<<blocks((size_t)N * 192), TB, 0, stream>>>(
        m1b, ab1, aa1, PIN(38), PIN(39), f1b, out, N);

    // ======================= LAYER 1 =======================
    // adapter-before (scalars only)
    gemm(f0b, PIN(57), d16, nullptr, N, 64, 16, 1.0f, 1, stream);
    gemm(d16, PIN(58), ab0, nullptr, N, 16, 64, 1.0f, 0, stream);
    k_bn_stats0<<<64, 256, 0, stream>>>(ab0, N, s1, s2);
    k_bn_apply0<<<blocks((size_t)N * 64), TB, 0, stream>>>(ab0, PIN(59), PIN(60), s1, s2, ab0, N);

    // interaction
    gemm(f0b, PIN(40), x0up, nullptr, N, 64, 64, 1.0f, 0, stream);
    gemm(f1b, PIN(41), x1up, nullptr, 3 * N, 64, 64, 1.0f, 0, stream);
    (void)hipMemsetAsync(seg0, 0, (size_t)N * 64 * sizeof(float), stream);
    (void)hipMemsetAsync(seg1, 0, (size_t)N * 192 * sizeof(float), stream);
    for (int c = 0; c < 8; ++c) {
        int base = c * CH;
        gemm(ef + (size_t)base * 8, PIN(42), h1, PIN(43), CH, 8, 64, 1.0f, 1, stream);
        gemm(h1, PIN(44), h2, PIN(45), CH, 64, 64, 1.0f, 1, stream);
        gemm(h2, PIN(46), Wc, nullptr, CH, 64, 256, 1.0f, 0, stream);
        k_scatter_l1<<<blocks((size_t)CH * 64), TB, 0, stream>>>(
            Wc, x0up, x1up, Y1, send, recv, seg0, seg1, base, CH);
    }
    gemm(seg0, PIN(47), a0, nullptr, N, 64, 64, invAvg, 0, stream);   // no mix
    gemm(seg1, PIN(48), a1, nullptr, 3 * N, 64, 64, invAvg, 0, stream);

    // message (scalars only output)
    k_msg_elem<<<blocks((size_t)N * 64), TB, 0, stream>>>(
        a0, a1, z, PIN(49), PIN(50), PIN(51), nullptr, nullptr, t0, nullptr, N);
    gemm(t0, PIN(52), m0b, nullptr, N, 64, 64, 1.0f, 0, stream);

    // node update with residual skip: u0 = seg0
    gemm(m0b, PIN(53), seg0, nullptr, N, 64, 64, 1.0f, 0, stream);
    k_skip_gemv<<<N, 64, 0, stream>>>(f0b, PIN(54), z, seg0, 1, 1);

    // layer BN (scalars): b0 -> m0b
    k_bn_stats0<<<64, 256, 0, stream>>>(seg0, N, s1, s2);
    k_bn_apply0<<<blocks((size_t)N * 64), TB, 0, stream>>>(seg0, PIN(55), PIN(56), s1, s2, m0b, N);

    // adapter-after (scalars)
    gemm(seg0, PIN(61), d16, nullptr, N, 64, 16, 1.0f, 1, stream);
    gemm(d16, PIN(62), aa0, nullptr, N, 16, 64, 1.0f, 0, stream);
    k_bn_stats0<<<64, 256, 0, stream>>>(aa0, N, s1, s2);
    k_bn_apply0<<<blocks((size_t)N * 64), TB, 0, stream>>>(aa0, PIN(63), PIN(64), s1, s2, aa0, N);

    // combine -> output cols [256,320)
    k_combine0<<<blocks((size_t)N * 64), TB, 0, stream>>>(
        m0b, ab0, aa0, PIN(65), PIN(66), nullptr, out, 256, N);
#undef PIN
}